// GraphBuilder_85177791415026
// MI455X (gfx1250) — compile-verified
//
#include <hip/hip_runtime.h>
#include <math.h>

typedef float v2f __attribute__((ext_vector_type(2)));
typedef float v8f __attribute__((ext_vector_type(8)));

#define B_ 8
#define T_ 16
#define N_ 256
#define S_ 3

// ---------------------------------------------------------------------------
// Kernel A: greedy group assignment per batch on last-timestep distances.
// One block (256 threads) per batch; sequential over pivot i with LDS flags.
// dist <= 2.0  <=>  sq <= 4.0 (sqrt monotone, 2.0 exact), so no sqrt needed.
// ---------------------------------------------------------------------------
__global__ __launch_bounds__(256) void group_kernel(const float* __restrict__ pos,
                                                    float* __restrict__ out_groups) {
    const int b = blockIdx.x;
    const int j = threadIdx.x;
    const float* p = pos + ((size_t)(b * T_ + (T_ - 1)) * N_) * 2;
    const float xj = p[2 * j + 0];
    const float yj = p[2 * j + 1];

    __shared__ float sx[N_], sy[N_];
    __shared__ int sAss[N_];
    sx[j] = xj; sy[j] = yj; sAss[j] = 0;
    __syncthreads();

    int  myGroup = 0;
    bool myAss   = false;
    int  cur     = 0;

    for (int i = 0; i < N_; ++i) {
        const int iAss = sAss[i];      // uniform across block
        __syncthreads();               // all reads of sAss[i] done before writes
        if (!iAss) {
            const float dx = sx[i] - xj;
            const float dy = sy[i] - yj;
            const float sq = dx * dx + dy * dy;
            if (!myAss && (sq <= 4.0f) && (j >= i)) {
                myGroup = cur;
                myAss   = true;
                sAss[j] = 1;
            }
            cur += 1;                  // uniform update, matches reference 'cur'
        }
        __syncthreads();
    }
    out_groups[b * N_ + j] = (float)myGroup;
}

// ---------------------------------------------------------------------------
// Kernel B: one wave per 16x16 tile per (b,t).
// Cross term via V_WMMA_F32_16X16X4_F32 (K=2 padded to 4 with zeros).
// A layout (16x4 f32): lanes 0-15 hold row M=lane (V0=K0, V1=K1); lanes 16-31
// hold K2,K3 (zeros). B layout (4x16): V0 = K0 row over lanes 0-15, V1 = K1;
// lanes 16-31 (K2,K3) zero. C/D: VGPR v -> M = v + (lane>=16 ? 8 : 0),
// N = lane%16.
// Position distances only feed threshold compares -> compare squared values.
// ---------------------------------------------------------------------------
__global__ __launch_bounds__(256) void adj_kernel(const float* __restrict__ pos,
                                                  const float* __restrict__ disp,
                                                  const float* __restrict__ groups,
                                                  float* __restrict__ adj,
                                                  float* __restrict__ intra) {
    const int lane = threadIdx.x & 31;
    const int wave = threadIdx.x >> 5;
    const int id = blockIdx.x * 8 + wave;           // tile id in [0, B*T*256)
    const int tj = id & 15;
    const int ti = (id >> 4) & 15;
    const int t  = (id >> 8) & 15;
    const int b  = id >> 12;

    const int i0 = ti * 16;
    const int j0 = tj * 16;
    const int l15 = lane & 15;

    const float2* pp = (const float2*)(pos  + ((size_t)(b * T_ + t) * N_) * 2);
    const float2* dp = (const float2*)(disp + ((size_t)(b * T_ + t) * N_) * 2);
    const float*  gr = groups + b * N_;

    v2f pa = {0.0f, 0.0f}, da = {0.0f, 0.0f};   // A tiles (rows)
    v2f pb = {0.0f, 0.0f}, db = {0.0f, 0.0f};   // B tiles (cols)
    float prow = 0.0f, drow = 0.0f;             // row norms (valid lanes 0-15)
    float pcol = 0.0f, dcol = 0.0f;             // col norms (valid lanes 0-15)
    float growf = 0.0f, gcolf = 0.0f;           // group labels (valid lanes 0-15)

    if (lane < 16) {
        const float2 ra = pp[i0 + lane];
        const float2 ca = pp[j0 + lane];
        const float2 rd = dp[i0 + lane];
        const float2 cd = dp[j0 + lane];
        pa.x = ra.x; pa.y = ra.y;
        pb.x = ca.x; pb.y = ca.y;
        da.x = rd.x; da.y = rd.y;
        db.x = cd.x; db.y = cd.y;
        prow = ra.x * ra.x + ra.y * ra.y;
        pcol = ca.x * ca.x + ca.y * ca.y;
        drow = rd.x * rd.x + rd.y * rd.y;
        dcol = cd.x * cd.x + cd.y * cd.y;
        growf = gr[i0 + lane];
        gcolf = gr[j0 + lane];
    }

    // Broadcast column-associated values so every lane has its column's data.
    const float pnj = __shfl(pcol, l15);
    const float dnj = __shfl(dcol, l15);
    const float gj  = __shfl(gcolf, l15);

    const v8f zero = {};
    // D = A x B + 0 : 16x16 Gram cross terms (full wave, EXEC all ones here).
    v8f pdot = __builtin_amdgcn_wmma_f32_16x16x4_f32(
        false, pa, false, pb, (short)0, zero, false, false);
    v8f ddot = __builtin_amdgcn_wmma_f32_16x16x4_f32(
        false, da, false, db, (short)0, zero, false, false);

    const int hi8 = (lane >= 16) ? 8 : 0;
    const float scales2[S_] = {0.25f, 1.0f, 4.0f};   // squared thresholds

    // Uniform output base pointers per (scale, buffer); per-element offsets are
    // small 32-bit values -> cheap saddr+voffset addressing on the stores.
    const size_t btBase = ((size_t)(b * S_) * T_ + t) << 16;
    float* __restrict__ adjP[S_];
    float* __restrict__ intraP[S_];
#pragma unroll
    for (int s = 0; s < S_; ++s) {
        const size_t off = btBase + ((size_t)(s * T_) << 16);
        adjP[s]   = adj   + off;
        intraP[s] = intra + off;
    }

    const int colOff = j0 + l15;

#pragma unroll
    for (int v = 0; v < 8; ++v) {
        const int m = v + hi8;                     // row within tile, 0..15
        const float pni = __shfl(prow, m);         // row norm from lane m
        const float dni = __shfl(drow, m);
        const float gi  = __shfl(growf, m);

        // squared position distance (clamped): only used for compares
        const float pd2 = fmaxf(pni + pnj - 2.0f * pdot[v], 0.0f);
        // displacement distance: raw v_sqrt_f32 + v_rcp_f32 (branch-free)
        const float dd2   = fmaxf(dni + dnj - 2.0f * ddot[v], 0.0f);
        const float ddist = __builtin_amdgcn_sqrtf(dd2);
        const float w     = __builtin_amdgcn_rcpf(ddist + 1e-6f);

        const float gm = (gi == gj) ? 1.0f : 0.0f;
        const int elem = (i0 + m) * N_ + colOff;

#pragma unroll
        for (int s = 0; s < S_; ++s) {
            const float a = (pd2 <= scales2[s]) ? w : 0.0f;
            __builtin_nontemporal_store(a,      adjP[s]   + elem);
            __builtin_nontemporal_store(a * gm, intraP[s] + elem);
        }
    }
}

// ---------------------------------------------------------------------------
extern "C" void kernel_launch(void* const* d_in, const int* in_sizes, int n_in,
                              void* d_out, int out_size, void* d_ws, size_t ws_size,
                              hipStream_t stream) {
    (void)in_sizes; (void)n_in; (void)out_size; (void)d_ws; (void)ws_size;

    const float* pos  = (const float*)d_in[0];
    const float* disp = (const float*)d_in[1];
    float* out = (float*)d_out;

    const size_t adjN = (size_t)B_ * S_ * T_ * N_ * N_;   // 25,165,824
    float* adj     = out;
    float* grp_out = out + adjN;
    float* intra   = out + adjN + (size_t)B_ * N_;

    // 1) group assignment (writes group labels as floats into d_out slot)
    group_kernel<<<B_, N_, 0, stream>>>(pos, grp_out);

    // 2) adjacency + intra-adjacency: 8 waves/block, one 16x16 tile per wave
    const int tiles = B_ * T_ * (N_ / 16) * (N_ / 16);    // 32768
    adj_kernel<<<tiles / 8, 256, 0, stream>>>(pos, disp, grp_out, adj, intra);
}